// DHGNN_89739046683505
// MI455X (gfx1250) — compile-verified
//
#include <hip/hip_runtime.h>
#include <hip/hip_bf16.h>

// CDNA5 / gfx1250, wave32. All matrix math via V_WMMA_F32_16X16X4_F32 (fp32 WMMA).

typedef float v2f __attribute__((ext_vector_type(2)));
typedef float v8f __attribute__((ext_vector_type(8)));

#define WMMA_F32(a, b, c) \
  __builtin_amdgcn_wmma_f32_16x16x4_f32(false, (a), false, (b), (short)0, (c), false, false)

// ---------------------------------------------------------------------------
// Accumulate acc += W_tile * X_tile over Cs channels of one source tensor.
// wp walks W row (o0+l) starting at the current channel (+koff within chunk);
// xp walks channels of X at fixed column n0+l. Pure pointer increments so the
// backend strength-reduces addressing and pipelines the loads.
// ---------------------------------------------------------------------------
__device__ __forceinline__ void gemm_acc(v8f& acc, const float*& wp,
                                         const float* xp, int Cs, int Nn) {
  const size_t xstep = (size_t)4 * Nn;
#pragma unroll 4
  for (int k0 = 0; k0 < Cs; k0 += 4) {
    v2f a, bb;
    a.x = wp[0];
    a.y = wp[1];
    bb.x = xp[0];
    bb.y = xp[Nn];
    acc = WMMA_F32(a, bb, acc);
    wp += 4;
    xp += xstep;
  }
}

// ---------------------------------------------------------------------------
// Generic fp32 WMMA GEMM:  Y[b,o,n] = sum_i W[o,i] * Xsel[b,i,n] + bias[o]
// Xsel: channels [0,ksplit) from X0, [ksplit,I) from X1 (concat conv).
// One wave computes a 16(o) x 16(n) tile; grid = (Nn/16, O/16, 4).
// ---------------------------------------------------------------------------
__global__ void gemm_f32_kernel(float* __restrict__ Y,
                                const float* __restrict__ W,
                                const float* __restrict__ bias,
                                const float* __restrict__ X0,
                                const float* __restrict__ X1,
                                int O, int I, int ksplit, int Nn) {
  const int lane = threadIdx.x;
  const bool hi = lane >= 16;
  const int l = lane & 15;
  const int koff = hi ? 2 : 0;
  const int n0 = blockIdx.x * 16;
  const int o0 = blockIdx.y * 16;
  const int b  = blockIdx.z;
  v8f acc = {};

  const float* wp = W + (size_t)(o0 + l) * I + koff;
  const float* xp0 = X0 + ((size_t)b * ksplit + koff) * Nn + n0 + l;
  gemm_acc(acc, wp, xp0, ksplit, Nn);
  if (X1) {
    const int C1 = I - ksplit;
    const float* xp1 = X1 + ((size_t)b * C1 + koff) * Nn + n0 + l;
    gemm_acc(acc, wp, xp1, C1, Nn);
  }

  const int rb = hi ? 8 : 0;
#pragma unroll
  for (int r = 0; r < 8; ++r) {
    const int o = o0 + r + rb;
    Y[((size_t)b * O + o) * Nn + n0 + l] = acc[r] + bias[o];
  }
}

// ---------------------------------------------------------------------------
// Multi-head attention, one wave per (b, h, 16-query tile).
// q:(4,128,Nq) k,v:(4,128,M) channel layout ch = c*H + h  (H=4, d=32).
// Full 16xM score block kept in LDS (<=128KB; CDNA5 has 320KB/WGP).
// Writes raw msg (pre-Wm) and accumulates prob column sums into sp (atomics).
// ---------------------------------------------------------------------------
__global__ void attn_kernel(const float* __restrict__ q,
                            const float* __restrict__ kbuf,
                            const float* __restrict__ vbuf,
                            float* __restrict__ msg,
                            float* __restrict__ sp,
                            int Nq, int M, int crossSwap) {
  extern __shared__ float smem[];       // 16 * M floats: scores -> probs
  __shared__ float s_red[2][16];
  const int lane = threadIdx.x;
  const bool hi = lane >= 16;
  const int l = lane & 15;
  const int n0 = blockIdx.x * 16;
  const int h  = blockIdx.y;
  const int b  = blockIdx.z;
  const int H = 4;
  const size_t qbase = (size_t)b * 128 * Nq;

  // Preload Q fragments (A operand, WMMA layout) for all 8 K-chunks of c-dim.
  v2f af[8];
#pragma unroll
  for (int kc = 0; kc < 8; ++kc) {
    const int ka = kc * 4 + (hi ? 2 : 0);
    af[kc].x = q[qbase + (size_t)(ka * H + h) * Nq + n0 + l];
    af[kc].y = q[qbase + (size_t)((ka + 1) * H + h) * Nq + n0 + l];
  }
  // Per-chunk K row pointers (loop-invariant; inner loop just adds m0).
  const float* kp0[8];
  const float* kp1[8];
#pragma unroll
  for (int kc = 0; kc < 8; ++kc) {
    const int ka = kc * 4 + (hi ? 2 : 0);
    kp0[kc] = kbuf + (size_t)b * 128 * M + (size_t)(ka * H + h) * M + l;
    kp1[kc] = kbuf + (size_t)b * 128 * M + (size_t)((ka + 1) * H + h) * M + l;
  }
  const float scale = 0.17677669529663687f;   // 1/sqrt(32)

  // Pass 1: scores = (Q^T K) * scale  -> LDS
  for (int m0 = 0; m0 < M; m0 += 16) {
    v8f acc = {};
#pragma unroll
    for (int kc = 0; kc < 8; ++kc) {
      v2f bf;
      bf.x = kp0[kc][m0];
      bf.y = kp1[kc][m0];
      acc = WMMA_F32(af[kc], bf, acc);
    }
    const int rb = hi ? 8 : 0;
#pragma unroll
    for (int r = 0; r < 8; ++r)
      smem[(size_t)(r + rb) * M + m0 + l] = acc[r] * scale;
  }
  __syncthreads();

  // Softmax over each of 16 rows; lane pair (l, l+16) splits the m-range.
  // Vectorized (float4) LDS traffic: M is a multiple of 16, halves of 8.
  {
    const int halfSel = hi ? 1 : 0;
    const int mq = (M >> 3);               // quarter-count per half (float4s)
    float4* row4 = (float4*)(smem + (size_t)l * M) + halfSel * mq;
    float pmax = -3.0e38f;
    for (int m = 0; m < mq; ++m) {
      const float4 v = row4[m];
      pmax = fmaxf(pmax, fmaxf(fmaxf(v.x, v.y), fmaxf(v.z, v.w)));
    }
    s_red[halfSel][l] = pmax;
    __syncthreads();
    const float rmax = fmaxf(s_red[0][l], s_red[1][l]);
    float psum = 0.f;
    for (int m = 0; m < mq; ++m) {
      float4 v = row4[m];
      v.x = __expf(v.x - rmax);
      v.y = __expf(v.y - rmax);
      v.z = __expf(v.z - rmax);
      v.w = __expf(v.w - rmax);
      row4[m] = v;
      psum += (v.x + v.y) + (v.z + v.w);
    }
    __syncthreads();
    s_red[halfSel][l] = psum;
    __syncthreads();
    const float inv = 1.0f / (s_red[0][l] + s_red[1][l]);
    for (int m = 0; m < mq; ++m) {
      float4 v = row4[m];
      v.x *= inv; v.y *= inv; v.z *= inv; v.w *= inv;
      row4[m] = v;
    }
  }
  __syncthreads();

  // Prob column sums for pooling: sp[b', m] += (1/H) * sum_n prob[n, m]
  {
    const int bOut = crossSwap ? ((b + 2) & 3) : b;
    for (int m = lane * 4; m < M; m += 128) {
      float4 cs = make_float4(0.f, 0.f, 0.f, 0.f);
#pragma unroll
      for (int r = 0; r < 16; ++r) {
        const float4 v = *(const float4*)(smem + (size_t)r * M + m);
        cs.x += v.x; cs.y += v.y; cs.z += v.z; cs.w += v.w;
      }
      atomicAdd(&sp[(size_t)bOut * M + m + 0], 0.25f * cs.x);
      atomicAdd(&sp[(size_t)bOut * M + m + 1], 0.25f * cs.y);
      atomicAdd(&sp[(size_t)bOut * M + m + 2], 0.25f * cs.z);
      atomicAdd(&sp[(size_t)bOut * M + m + 3], 0.25f * cs.w);
    }
  }
  __syncthreads();

  // Pass 2: msg = P @ V^T  (16 n  x  32 c), P streamed from LDS.
  const size_t mbase = (size_t)b * 128 * Nq;
#pragma unroll
  for (int c0 = 0; c0 < 32; c0 += 16) {
    v8f acc = {};
    const float* ap = smem + (size_t)l * M + (hi ? 2 : 0);
    const float* vp = vbuf + (size_t)b * 128 * M +
                      (size_t)((c0 + l) * H + h) * M + (hi ? 2 : 0);
#pragma unroll 4
    for (int m0 = 0; m0 < M; m0 += 4) {
      v2f a, bf;
      a.x = ap[0];
      a.y = ap[1];
      bf.x = vp[0];
      bf.y = vp[1];
      acc = WMMA_F32(a, bf, acc);
      ap += 4;
      vp += 4;
    }
    const int rb = hi ? 8 : 0;
#pragma unroll
    for (int r = 0; r < 8; ++r)
      msg[mbase + (size_t)((c0 + l) * H + h) * Nq + n0 + r + rb] = acc[r];
  }
}

// ---------------------------------------------------------------------------
// Per-channel mean/var over (B=4, N) for batchnorm. One block per channel.
// ---------------------------------------------------------------------------
__global__ void stats_kernel(const float* __restrict__ Hh,
                             float* __restrict__ mu, float* __restrict__ var,
                             int C, int Nn) {
  __shared__ float ssum[256], ssq[256];
  const int o = blockIdx.x, t = threadIdx.x;
  float s = 0.f, sq = 0.f;
  const int total = 4 * Nn;
  for (int i = t; i < total; i += 256) {
    const int b = i / Nn, n = i - b * Nn;
    const float x = Hh[((size_t)b * C + o) * Nn + n];
    s += x; sq += x * x;
  }
  ssum[t] = s; ssq[t] = sq;
  __syncthreads();
  for (int st = 128; st > 0; st >>= 1) {
    if (t < st) { ssum[t] += ssum[t + st]; ssq[t] += ssq[t + st]; }
    __syncthreads();
  }
  if (t == 0) {
    const float m = ssum[0] / (float)total;
    mu[o] = m;
    var[o] = ssq[0] / (float)total - m * m;
  }
}

__global__ void bn_relu_kernel(float* __restrict__ Hh,
                               const float* __restrict__ mu, const float* __restrict__ var,
                               const float* __restrict__ gamma, const float* __restrict__ beta,
                               int C, int Nn) {
  const size_t i = (size_t)blockIdx.x * blockDim.x + threadIdx.x;
  const size_t total = (size_t)4 * C * Nn;
  if (i >= total) return;
  const int o = (int)((i / Nn) % C);
  float x = Hh[i];
  x = (x - mu[o]) * rsqrtf(var[o] + 1.0e-3f) * gamma[o] + beta[o];
  Hh[i] = fmaxf(x, 0.f);
}

// ---------------------------------------------------------------------------
// Exact top-k (JAX tie-break: lower index wins), output indices sorted asc,
// mapped through oldIdx (nullptr => identity). One block per batch row.
// ---------------------------------------------------------------------------
__global__ void topk_kernel(const float* __restrict__ sp,
                            const int* __restrict__ oldIdx,
                            int* __restrict__ newIdx, int M, int k) {
  __shared__ int flags[2048];
  const int b = blockIdx.x;
  const float* row = sp + (size_t)b * M;
  for (int m = threadIdx.x; m < M; m += blockDim.x) {
    const float v = row[m];
    int rank = 0;
    for (int j = 0; j < M; ++j) {
      const float vj = row[j];
      rank += (vj > v) || (vj == v && j < m);
    }
    flags[m] = (rank < k) ? 1 : 0;
  }
  __syncthreads();
  if (threadIdx.x == 0) {
    int c = 0;
    for (int m = 0; m < M; ++m)
      if (flags[m]) { newIdx[b * 2048 + c] = oldIdx ? oldIdx[b * 2048 + m] : m; ++c; }
  }
}

// src[b,c,j] = X[bsel, c, idx[bsel][j]]  (bsel batch-swapped for cross layers)
__global__ void build_src_kernel(float* __restrict__ S, const float* __restrict__ X,
                                 const int* __restrict__ idx, int Msrc, int swap) {
  const size_t i = (size_t)blockIdx.x * blockDim.x + threadIdx.x;
  const size_t total = (size_t)4 * 128 * Msrc;
  if (i >= total) return;
  const int j = (int)(i % Msrc);
  const int c = (int)((i / Msrc) % 128);
  const int b = (int)(i / ((size_t)128 * Msrc));
  const int bs = swap ? ((b + 2) & 3) : b;
  const int col = idx ? idx[bs * 2048 + j] : j;
  S[i] = X[((size_t)bs * 128 + c) * 2048 + col];
}

__global__ void copy_kernel(float* __restrict__ d, const float* __restrict__ s, size_t n) {
  const size_t i = (size_t)blockIdx.x * blockDim.x + threadIdx.x;
  if (i < n) d[i] = s[i];
}
__global__ void add_kernel(float* __restrict__ d, const float* __restrict__ s, size_t n) {
  const size_t i = (size_t)blockIdx.x * blockDim.x + threadIdx.x;
  if (i < n) d[i] += s[i];
}
__global__ void zero_kernel(float* __restrict__ d, size_t n) {
  const size_t i = (size_t)blockIdx.x * blockDim.x + threadIdx.x;
  if (i < n) d[i] = 0.f;
}

// ---------------------------------------------------------------------------
extern "C" void kernel_launch(void* const* d_in, const int* in_sizes, int n_in,
                              void* d_out, int out_size, void* d_ws, size_t ws_size,
                              hipStream_t stream) {
  (void)in_sizes; (void)n_in; (void)out_size; (void)ws_size;
  const int N = 2048;
  const size_t PLANE = (size_t)4 * 128 * N;      // 1,048,576 floats
  const size_t HALF  = PLANE / 2;                // one (2,128,2048) slab

  float* ws  = (float*)d_ws;
  float* X   = ws;                // persistent: concat(desc0, desc1)
  float* SRC = X   + PLANE;       // src set; reused as raw msg
  float* Q   = SRC + PLANE;       // q; reused as msg after Wm
  float* Kb  = Q   + PLANE;       // k; reused as delta
  float* Vb  = Kb  + PLANE;
  float* Hh  = Vb  + PLANE;       // (4,256,2048)
  float* mu  = Hh  + 2 * PLANE;
  float* var = mu  + 256;
  float* spS = var + 256;         // (4,2048)
  float* spC = spS + 4 * 2048;
  int* idxS0 = (int*)(spC + 4 * 2048);
  int* idxS1 = idxS0 + 4 * 2048;
  int* idxC0 = idxS1 + 4 * 2048;
  int* idxC1 = idxC0 + 4 * 2048;

  // Params: 6 layers x 14 leaves, JAX pytree order (dict keys sorted).
  const float* P[6][14];
  for (int L = 0; L < 6; ++L)
    for (int j = 0; j < 14; ++j)
      P[L][j] = (const float*)d_in[2 + L * 14 + j];
  enum { iW1 = 0, iW2, iWk, iWm, iWq, iWv, ib1, ib2, iBeta, ibk, ibm, ibq, ibv, iGamma };

  // X = concat(desc0, desc1) along batch
  copy_kernel<<<(unsigned)((HALF + 255) / 256), 256, 0, stream>>>(X, (const float*)d_in[0], HALF);
  copy_kernel<<<(unsigned)((HALF + 255) / 256), 256, 0, stream>>>(X + HALF, (const float*)d_in[1], HALF);

  int MspS = 0, MspC = 0;
  const int* idxScur = nullptr; const int* idxCcur = nullptr;
  int sFlip = 0, cFlip = 0;
  float* out = (float*)d_out;
  int slot = 0;

  for (int L = 0; L < 6; ++L) {
    const bool cross = (L & 1) != 0;
    const int ps = (L < 2) ? 0 : 2;
    int Msrc = N;
    const int* gidx = nullptr;

    if (ps > 0) {
      float* sp = cross ? spC : spS;
      const int Mp = cross ? MspC : MspS;
      int k = N / ps; if (k < 128) k = 128; if (k > N) k = N; if (k > Mp) k = Mp;
      int* nidx;
      if (cross) { nidx = cFlip ? idxC1 : idxC0; cFlip ^= 1; }
      else       { nidx = sFlip ? idxS1 : idxS0; sFlip ^= 1; }
      topk_kernel<<<4, 256, 0, stream>>>(sp, cross ? idxCcur : idxScur, nidx, Mp, k);
      if (cross) idxCcur = nidx; else idxScur = nidx;
      gidx = nidx;
      Msrc = k;
    }

    build_src_kernel<<<(unsigned)(((size_t)4 * 128 * Msrc + 255) / 256), 256, 0, stream>>>(
        SRC, X, gidx, Msrc, cross ? 1 : 0);

    // q/k/v projections (fp32 WMMA GEMMs)
    gemm_f32_kernel<<<dim3(N / 16, 8, 4), 32, 0, stream>>>(
        Q,  P[L][iWq], P[L][ibq], X,   nullptr, 128, 128, 128, N);
    gemm_f32_kernel<<<dim3(Msrc / 16, 8, 4), 32, 0, stream>>>(
        Kb, P[L][iWk], P[L][ibk], SRC, nullptr, 128, 128, 128, Msrc);
    gemm_f32_kernel<<<dim3(Msrc / 16, 8, 4), 32, 0, stream>>>(
        Vb, P[L][iWv], P[L][ibv], SRC, nullptr, 128, 128, 128, Msrc);

    float* sp = cross ? spC : spS;
    zero_kernel<<<32, 256, 0, stream>>>(sp, 4 * 2048);

    attn_kernel<<<dim3(N / 16, 4, 4), 32, (size_t)16 * Msrc * sizeof(float), stream>>>(
        Q, Kb, Vb, SRC, sp, N, Msrc, cross ? 1 : 0);
    if (cross) MspC = Msrc; else MspS = Msrc;

    // msg = Wm @ msg_raw + bm   (SRC -> Q)
    gemm_f32_kernel<<<dim3(N / 16, 8, 4), 32, 0, stream>>>(
        Q, P[L][iWm], P[L][ibm], SRC, nullptr, 128, 128, 128, N);
    // h = W1 @ concat(X, msg) + b1
    gemm_f32_kernel<<<dim3(N / 16, 16, 4), 32, 0, stream>>>(
        Hh, P[L][iW1], P[L][ib1], X, Q, 256, 256, 128, N);
    stats_kernel<<<256, 256, 0, stream>>>(Hh, mu, var, 256, N);
    bn_relu_kernel<<<(unsigned)((2 * PLANE + 255) / 256), 256, 0, stream>>>(
        Hh, mu, var, P[L][iGamma], P[L][iBeta], 256, N);
    // delta = W2 @ h + b2   (-> Kb)
    gemm_f32_kernel<<<dim3(N / 16, 8, 4), 32, 0, stream>>>(
        Kb, P[L][iW2], P[L][ib2], Hh, nullptr, 128, 256, 256, N);
    add_kernel<<<(unsigned)((PLANE + 255) / 256), 256, 0, stream>>>(X, Kb, PLANE);

    if (cross) {
      copy_kernel<<<(unsigned)((HALF + 255) / 256), 256, 0, stream>>>(
          out + (size_t)slot * HALF, X, HALF);
      copy_kernel<<<(unsigned)((HALF + 255) / 256), 256, 0, stream>>>(
          out + (size_t)(3 + slot) * HALF, X + HALF, HALF);
      ++slot;
    }
  }
}